// Shift_DWConv_40132174414255
// MI455X (gfx1250) — compile-verified
//
#include <hip/hip_runtime.h>
#include <stdint.h>

// Shapes (compile-time constants from the reference)
#define C_DIM 768
#define B_DIM 16
#define H_DIM 64
#define W_DIM 64
#define N_DIM (H_DIM * W_DIM)

// Tiling
#define TR 8              // output rows per block
#define CB 16             // channels per block (4 float4 quads)
#define LROWS (TR + 2)    // staged rows incl. halo

typedef __attribute__((ext_vector_type(4))) float v4f;  // native vector for NT store

// DeepShift fixed-point quantize: floor(x * 2^16) / 2^16, clip to [-2^15, 2^15-1]
// med3 == clamp for finite inputs; 1 VALU op instead of min+max.
__device__ __forceinline__ float q16(float v) {
    float t = floorf(v * 65536.0f) * (1.0f / 65536.0f);
    return __builtin_amdgcn_fmed3f(t, -32768.0f, 32767.0f);
}

// ---------------------------------------------------------------------------
// Prep: decode power-of-two weights into [tap][C] layout + quantized bias
// ---------------------------------------------------------------------------
__global__ void __launch_bounds__(256) shift_prep_kernel(
        const float* __restrict__ shift,   // [C,1,3,3] -> c*9+tap
        const float* __restrict__ sgn,     // [C,1,3,3]
        const float* __restrict__ bias,    // [C]
        float* __restrict__ wq,            // [9][C]
        float* __restrict__ bq) {          // [C]
    int i = blockIdx.x * 256 + threadIdx.x;
    if (i < C_DIM * 9) {
        int c = i / 9, tap = i % 9;
        float s  = fminf(fmaxf(shift[i], -14.0f), 0.0f);
        float sr = rintf(s);                       // jnp.round == RNE rint
        float g  = rintf(sgn[i]);
        float sg = (g > 0.0f) ? 1.0f : ((g < 0.0f) ? -1.0f : 0.0f);
        wq[tap * C_DIM + c] = ldexpf(sg, (int)sr); // exact +/- 2^sr (or 0)
    }
    if (i < C_DIM) bq[i] = q16(bias[i]);
}

// Raw float4 read from the LDS tile (data already quantized in place)
__device__ __forceinline__ float4 ldsr(const float* tile, int lr, int col, int q) {
    return *(const float4*)&tile[((lr * W_DIM + col) * CB) + (q << 2)];
}

#define FMA4(o, wv, av)                         \
    o.x = fmaf((wv).x, (av).x, o.x);            \
    o.y = fmaf((wv).y, (av).y, o.y);            \
    o.z = fmaf((wv).z, (av).z, o.z);            \
    o.w = fmaf((wv).w, (av).w, o.w)

// ---------------------------------------------------------------------------
// Main: async-LDS staged depthwise 3x3 stencil, channel-last
// grid = (H/TR, C/CB, B), block = 256 (8 waves of 32)
// ---------------------------------------------------------------------------
__global__ void __launch_bounds__(256) shift_dwconv_kernel(
        const float* __restrict__ x,       // [B, N, C]
        const float* __restrict__ wq,      // [9][C]
        const float* __restrict__ bq,      // [C]
        float* __restrict__ out) {         // [B, N, C]
    __shared__ float tile[LROWS * W_DIM * CB];   // 10*64*16*4 = 40 KB

    const int t  = threadIdx.x;
    const int q  = t & 3;          // channel quad 0..3
    const int w  = t >> 2;         // column 0..63
    const int r0 = blockIdx.x * TR;
    const int c0 = blockIdx.y * CB;
    const int b  = blockIdx.z;
    const int cq = c0 + q * 4;

    // ---- Stage LROWS input rows into LDS via async global->LDS B128 copies
    #pragma unroll
    for (int lr = 0; lr < LROWS; ++lr) {
        const int gr = r0 - 1 + lr;                       // uniform per block
        float* lslot = &tile[((lr * W_DIM + w) * CB) + q * 4];
        if ((unsigned)gr < (unsigned)H_DIM) {
            const float* gp = x + (((size_t)b * N_DIM + (size_t)gr * W_DIM + w) * C_DIM + cq);
            uint32_t loff = (uint32_t)(uintptr_t)lslot;   // low 32 bits = LDS offset
            uint64_t ga   = (uint64_t)(uintptr_t)gp;
            asm volatile("global_load_async_to_lds_b128 %0, %1, off"
                         :: "v"(loff), "v"(ga) : "memory");
        } else {
            float4 z = {0.0f, 0.0f, 0.0f, 0.0f};
            *(float4*)lslot = z;                          // H-boundary zero pad
        }
    }

    // ---- Per-quad weights (9 taps) and bias while copies are in flight
    float4 wt[9];
    #pragma unroll
    for (int tap = 0; tap < 9; ++tap)
        wt[tap] = *(const float4*)(wq + tap * C_DIM + cq);
    const float4 bias4 = *(const float4*)(bq + cq);

#if __has_builtin(__builtin_amdgcn_s_wait_asynccnt)
    __builtin_amdgcn_s_wait_asynccnt(0);
#else
    asm volatile("s_wait_asynccnt 0" ::: "memory");
#endif
    __syncthreads();

    // ---- Quantize staged activations in place, once per element
    // (each thread owns its own 10 slots: no cross-thread hazard)
    #pragma unroll
    for (int lr = 0; lr < LROWS; ++lr) {
        float4* p = (float4*)&tile[((lr * W_DIM + w) * CB) + q * 4];
        float4 v = *p;
        v.x = q16(v.x); v.y = q16(v.y); v.z = q16(v.z); v.w = q16(v.w);
        *p = v;
    }
    __syncthreads();

    // ---- Register sliding window: rows (ky=0,1,2) x cols (l,c,r) of float4
    const float4 z4 = {0.0f, 0.0f, 0.0f, 0.0f};
    const bool hasL = (w > 0);
    const bool hasR = (w < W_DIM - 1);

    float4 a0l = hasL ? ldsr(tile, 0, w - 1, q) : z4;
    float4 a0c =        ldsr(tile, 0, w,     q);
    float4 a0r = hasR ? ldsr(tile, 0, w + 1, q) : z4;
    float4 a1l = hasL ? ldsr(tile, 1, w - 1, q) : z4;
    float4 a1c =        ldsr(tile, 1, w,     q);
    float4 a1r = hasR ? ldsr(tile, 1, w + 1, q) : z4;

    #pragma unroll
    for (int r = 0; r < TR; ++r) {
        float4 a2l = hasL ? ldsr(tile, r + 2, w - 1, q) : z4;
        float4 a2c =        ldsr(tile, r + 2, w,     q);
        float4 a2r = hasR ? ldsr(tile, r + 2, w + 1, q) : z4;

        float4 o = bias4;
        FMA4(o, wt[0], a0l); FMA4(o, wt[1], a0c); FMA4(o, wt[2], a0r);
        FMA4(o, wt[3], a1l); FMA4(o, wt[4], a1c); FMA4(o, wt[5], a1r);
        FMA4(o, wt[6], a2l); FMA4(o, wt[7], a2c); FMA4(o, wt[8], a2r);

        const size_t oidx = ((size_t)b * N_DIM + (size_t)(r0 + r) * W_DIM + w) * C_DIM + cq;
        // Output is write-once: non-temporal store keeps L2 for input halo reuse.
        // Bit-cast to a native clang vector type (builtin rejects HIP_vector_type).
        v4f ov;
        ov.x = o.x; ov.y = o.y; ov.z = o.z; ov.w = o.w;
        __builtin_nontemporal_store(ov, (v4f*)(out + oidx));

        a0l = a1l; a0c = a1c; a0r = a1r;
        a1l = a2l; a1c = a2c; a1r = a2r;
    }
}

// ---------------------------------------------------------------------------
extern "C" void kernel_launch(void* const* d_in, const int* in_sizes, int n_in,
                              void* d_out, int out_size, void* d_ws, size_t ws_size,
                              hipStream_t stream) {
    (void)in_sizes; (void)n_in; (void)out_size; (void)ws_size;
    const float* x     = (const float*)d_in[0];
    const float* shift = (const float*)d_in[1];
    const float* sgn   = (const float*)d_in[2];
    const float* bias  = (const float*)d_in[3];
    // d_in[4] = H, d_in[5] = W (compile-time constants here)

    float* wq = (float*)d_ws;          // 9*768 floats
    float* bq = wq + 9 * C_DIM;        // 768 floats  (total ~30 KB of d_ws)
    float* out = (float*)d_out;

    shift_prep_kernel<<<(C_DIM * 9 + 255) / 256, 256, 0, stream>>>(shift, sgn, bias, wq, bq);

    dim3 grid(H_DIM / TR, C_DIM / CB, B_DIM);   // (8, 48, 16)
    shift_dwconv_kernel<<<grid, 256, 0, stream>>>(x, wq, bq, out);
}